// Policy_61512521613570
// MI455X (gfx1250) — compile-verified
//
#include <hip/hip_runtime.h>
#include <hip/hip_bf16.h>

// ---------------------------------------------------------------------------
// Problem constants (match reference)
// ---------------------------------------------------------------------------
#define N_NODES   50000
#define PER_GRAPH 100
#define NUM_GRAPHS (N_NODES / PER_GRAPH)   // 500
#define DEG       21
#define FORGET_K  16                        // keep 5 largest of 21
#define KEEP_K    (DEG - FORGET_K)          // 5
#define NODE_DIM  64
#define NUM_EDGES (N_NODES * DEG)           // 1,050,000

typedef __attribute__((ext_vector_type(2))) float v2f;
typedef __attribute__((ext_vector_type(8))) float v8f;

// ---------------------------------------------------------------------------
// Kernel 0: zero the per-graph accumulator region of the workspace
// ---------------------------------------------------------------------------
__global__ void k_init(float* __restrict__ bsum) {
    int i = blockIdx.x * blockDim.x + threadIdx.x;
    if (i < NUM_GRAPHS * 2) bsum[i] = 0.0f;
}

// ---------------------------------------------------------------------------
// Kernel 1: node projection  P[n, 0:4] = [x_n . W[0:64,0], x_n . W[0:64,1],
//                                         x_n . W[64:128,0], x_n . W[64:128,1]]
// via V_WMMA_F32_16X16X4_F32: one wave handles a 16-node tile, K=64 split
// into 16 k-steps of 4.  f32 end-to-end (precision matters for the top-k).
//
// A (16x4 f32) layout (ISA 7.12.2): lanes 0-15 = rows M=0..15,
//   VGPR0 = K = 4kk + 2*half, VGPR1 = K+1 (half = lane>=16).
// B (4x16 f32) layout mirrored: lane%16 = column N, VGPR0 = row K, VGPR1 = K+1.
// C/D (16x16 f32): VGPR r -> M = r + 8*half, N = lane%16.
// ---------------------------------------------------------------------------
__global__ void k_proj_wmma(const float* __restrict__ x,
                            const float* __restrict__ W,   // [128,2] row-major
                            float*       __restrict__ P) { // [N_NODES,4]
    const int tile = blockIdx.x;          // 0 .. 3124   (50000/16 tiles exactly)
    const int lane = threadIdx.x;         // 0 .. 31 (wave32)
    const int half = lane >> 4;           // 0 or 1
    const int m    = lane & 15;           // row (A) / column (B)
    const int nc   = m & 3;               // clamped B column (valid cols 0..3)
    const int wsel = (nc >> 1);           // 0 -> W_top, 1 -> W_bot
    const int wcol = (nc & 1);            // output channel 0/1

    const float* xrow = x + (tile * 16 + m) * NODE_DIM;
    const bool bvalid = (m < 4);

    v8f c = {};
#pragma unroll
    for (int kk = 0; kk < 16; ++kk) {
        const int k0 = kk * 4 + half * 2;        // K index of VGPR0
        v2f a;
        a.x = xrow[k0];
        a.y = xrow[k0 + 1];
        // Wc[k][nc] = W[(k + 64*wsel)*2 + wcol]
        const float w0 = W[(k0     + 64 * wsel) * 2 + wcol];
        const float w1 = W[(k0 + 1 + 64 * wsel) * 2 + wcol];
        v2f bv;
        bv.x = bvalid ? w0 : 0.0f;
        bv.y = bvalid ? w1 : 0.0f;
        c = __builtin_amdgcn_wmma_f32_16x16x4_f32(
                /*neg_a=*/false, a, /*neg_b=*/false, bv,
                /*c_mod=*/(short)0, c, /*reuse_a=*/false, /*reuse_b=*/false);
    }

    if (bvalid) {
        const int node0 = tile * 16 + half * 8;
#pragma unroll
        for (int r = 0; r < 8; ++r) {
            P[(node0 + r) * 4 + m] = c[r];
        }
    }
}

// ---------------------------------------------------------------------------
// Top-5-of-21 keep mask via 5 max-scans.  Ties prefer the HIGHER index:
// jax top_k(-att,16) prefers lower indices among ties for the *forgotten*
// set, so the kept set prefers higher indices.  (Ties are measure-zero here.)
// ---------------------------------------------------------------------------
__device__ __forceinline__ unsigned keep_top5(const float a[DEG]) {
    unsigned keep = 0;
#pragma unroll
    for (int t = 0; t < KEEP_K; ++t) {
        float best = -1.0f;   // sigmoid outputs are in (0,1)
        int   bi   = 0;
#pragma unroll
        for (int j = 0; j < DEG; ++j) {
            const bool taken = (keep >> j) & 1u;
            const float v = a[j];
            if (!taken && v >= best) { best = v; bi = j; }
        }
        keep |= (1u << bi);
    }
    return keep;
}

// ---------------------------------------------------------------------------
// Kernel 2: one thread per node.  att[e,c] = sigmoid(P[row,c] + P[col,2+c] + b[c])
// for the node's 21 edges, apply top-5 keep mask per channel, store masked att,
// atomically accumulate per-graph sums.
// ---------------------------------------------------------------------------
__global__ void k_att_forget(const float* __restrict__ P,
                             const int*   __restrict__ col,   // edge_index row 1
                             const float* __restrict__ bvec,  // [2]
                             float*       __restrict__ att,   // [E,2]
                             float*       __restrict__ bsum) {// [G,2]
    const int n = blockIdx.x * blockDim.x + threadIdx.x;
    if (n >= N_NODES) return;

    const float p0 = P[n * 4 + 0];
    const float p1 = P[n * 4 + 1];
    const float b0 = bvec[0];
    const float b1 = bvec[1];
    const int   base = n * DEG;

    float a0[DEG], a1[DEG];
#pragma unroll
    for (int j = 0; j < DEG; ++j) {
        const int cidx = col[base + j];
        const float z0 = p0 + P[cidx * 4 + 2] + b0;
        const float z1 = p1 + P[cidx * 4 + 3] + b1;
        a0[j] = 1.0f / (1.0f + __expf(-z0));
        a1[j] = 1.0f / (1.0f + __expf(-z1));
    }

    const unsigned keep0 = keep_top5(a0);
    const unsigned keep1 = keep_top5(a1);

    float s0 = 0.0f, s1 = 0.0f;
#pragma unroll
    for (int j = 0; j < DEG; ++j) {
        const float v0 = ((keep0 >> j) & 1u) ? a0[j] : 0.0f;
        const float v1 = ((keep1 >> j) & 1u) ? a1[j] : 0.0f;
        att[(base + j) * 2 + 0] = v0;
        att[(base + j) * 2 + 1] = v1;
        s0 += v0;
        s1 += v1;
    }

    const int g = n / PER_GRAPH;
    atomicAdd(&bsum[g * 2 + 0], s0);
    atomicAdd(&bsum[g * 2 + 1], s1);
}

// ---------------------------------------------------------------------------
// Kernel 3: per-edge normalization + weighted sum.
// out[e] = att[e,0]*aa[e,0]/max(S[g,0],1) + att[e,1]*aa[e,1]/max(S[g,1],1)
// ---------------------------------------------------------------------------
__global__ void k_normalize(const float* __restrict__ att,
                            const float* __restrict__ aa,
                            const float* __restrict__ bsum,
                            float*       __restrict__ out) {
    const int e = blockIdx.x * blockDim.x + threadIdx.x;
    if (e >= NUM_EDGES) return;
    const int g = e / (DEG * PER_GRAPH);          // == (e/21)/100
    const float s0 = fmaxf(bsum[g * 2 + 0], 1.0f);
    const float s1 = fmaxf(bsum[g * 2 + 1], 1.0f);
    const float o = att[e * 2 + 0] * aa[e * 2 + 0] / s0 +
                    att[e * 2 + 1] * aa[e * 2 + 1] / s1;
    out[e] = o;
}

// ---------------------------------------------------------------------------
// Host launcher.  d_in order: x, edge_index, actual_amount, W, b
// Workspace layout: P [50000*4 f32] | bsum [500*2 f32] | att [E*2 f32]
// ---------------------------------------------------------------------------
extern "C" void kernel_launch(void* const* d_in, const int* in_sizes, int n_in,
                              void* d_out, int out_size, void* d_ws, size_t ws_size,
                              hipStream_t stream) {
    const float* x  = (const float*)d_in[0];
    const int*   ei = (const int*)d_in[1];
    const float* aa = (const float*)d_in[2];
    const float* W  = (const float*)d_in[3];
    const float* b  = (const float*)d_in[4];
    float* out = (float*)d_out;

    // row half of edge_index is e/21 by construction; only col is needed.
    const int* col = ei + NUM_EDGES;

    char* ws = (char*)d_ws;
    float* P    = (float*)(ws);                                   // 800,000 B
    float* bsum = (float*)(ws + (size_t)N_NODES * 4 * sizeof(float));          // 4,000 B
    float* att  = (float*)(ws + (size_t)N_NODES * 4 * sizeof(float)
                               + (size_t)NUM_GRAPHS * 2 * sizeof(float));      // 8.4 MB

    (void)in_sizes; (void)n_in; (void)out_size; (void)ws_size;

    // 0) zero per-graph sums
    k_init<<<1, 1024, 0, stream>>>(bsum);

    // 1) WMMA node projection: 50000/16 = 3125 tiles, one wave32 each
    k_proj_wmma<<<N_NODES / 16, 32, 0, stream>>>(x, W, P);

    // 2) per-node attention + top-5 forgetting + graph sums
    k_att_forget<<<(N_NODES + 255) / 256, 256, 0, stream>>>(P, col, b, att, bsum);

    // 3) per-edge normalization
    k_normalize<<<(NUM_EDGES + 255) / 256, 256, 0, stream>>>(att, aa, bsum, out);
}